// MHLA2_29540785062640
// MI455X (gfx1250) — compile-verified
//
#include <hip/hip_runtime.h>
#include <hip/hip_bf16.h>

typedef __attribute__((ext_vector_type(2))) float v2f;
typedef __attribute__((ext_vector_type(8))) float v8f;

#define BQ     4
#define SEQ    2048
#define HEADS  16
#define DHEAD  64
#define DMODEL 1024
#define MROWS  (BQ * SEQ)   // 8192

static __device__ __forceinline__ v8f wmma_f32(v2f a, v2f b, v8f c) {
  // V_WMMA_F32_16X16X4_F32: D = A(16x4) * B(4x16) + C(16x16), full f32
  return __builtin_amdgcn_wmma_f32_16x16x4_f32(
      /*neg_a=*/false, a, /*neg_b=*/false, b,
      /*c_mod=*/(short)0, c, /*reuse_a=*/false, /*reuse_b=*/false);
}

// ---------------------------------------------------------------------------
// Pack W_{Q,K,V} [h, m, d] into WMMA-B interleaved layout:
//   Bp[(k>>1)*2048 + n*2 + (k&1)]  with  k = m, n = h*64 + d
// so each lane's B fragment pair {B[k,n], B[k+1,n]} is a contiguous float2.
// ---------------------------------------------------------------------------
__global__ __launch_bounds__(256) void pack_w_head(const float* __restrict__ W,
                                                   float* __restrict__ Bp) {
  int idx = blockIdx.x * 256 + threadIdx.x;       // 1M elements, coalesced read
  int h = idx >> 16;
  int m = (idx >> 6) & 1023;
  int d = idx & 63;
  int n = h * DHEAD + d;
  Bp[(size_t)(m >> 1) * (2 * DMODEL) + n * 2 + (m & 1)] = W[idx];
}

// W_O is [o, i] row-major; final GEMM uses B[k=i][n=o] = W_O[o][i], packed.
__global__ __launch_bounds__(256) void pack_wo(const float* __restrict__ W,
                                               float* __restrict__ Bp) {
  int idx = blockIdx.x * 256 + threadIdx.x;       // coalesced read
  int o = idx >> 10;
  int i = idx & 1023;
  Bp[(size_t)(i >> 1) * (2 * DMODEL) + o * 2 + (i & 1)] = W[idx];
}

// ---------------------------------------------------------------------------
// GEMM: C[8192,1024] = A[8192,1024] @ Bp(packed [512][1024][2])
// One wave computes a 16(M) x 64(N) strip: 4 accumulators share one A frag.
// All loads are global_load_b64; 4 wmma per 5 loads.
// ---------------------------------------------------------------------------
__global__ __launch_bounds__(256) void gemm_mn(const float* __restrict__ A,
                                               const float* __restrict__ Bp,
                                               float* __restrict__ C) {
  const int wave = blockIdx.x * 8 + (threadIdx.x >> 5);   // 8192 waves
  const int lane = threadIdx.x & 31;
  const int lr   = lane & 15;
  const int hi   = lane >> 4;              // K-pair select (0 or 1)
  const int khi  = hi << 1;                // K sub-offset (0 or 2)
  const int m0   = (wave >> 4) << 4;       // 512 m-tiles
  const int n0   = (wave & 15) << 6;       // 16 strips of 64 cols

  const float* arow = A + (size_t)(m0 + lr) * DMODEL;
  const float* bp   = Bp + (size_t)hi * (2 * DMODEL) + (n0 + lr) * 2;

  v8f acc0 = {}, acc1 = {}, acc2 = {}, acc3 = {};
#pragma unroll 4
  for (int k = 0; k < DMODEL; k += 4) {
    v2f a = *(const v2f*)(arow + k + khi);
    const float* bk = bp + (size_t)(k >> 1) * (2 * DMODEL);
    v2f b0 = *(const v2f*)(bk + 0);
    v2f b1 = *(const v2f*)(bk + 32);
    v2f b2 = *(const v2f*)(bk + 64);
    v2f b3 = *(const v2f*)(bk + 96);
    acc0 = wmma_f32(a, b0, acc0);
    acc1 = wmma_f32(a, b1, acc1);
    acc2 = wmma_f32(a, b2, acc2);
    acc3 = wmma_f32(a, b3, acc3);
  }

  const int ms = m0 + (hi << 3);
  float* cp = C + (size_t)ms * DMODEL + n0 + lr;
#pragma unroll
  for (int r = 0; r < 8; ++r) {
    cp[(size_t)r * DMODEL +  0] = acc0[r];
    cp[(size_t)r * DMODEL + 16] = acc1[r];
    cp[(size_t)r * DMODEL + 32] = acc2[r];
    cp[(size_t)r * DMODEL + 48] = acc3[r];
  }
}

// ---------------------------------------------------------------------------
// In-place softmax over each head-row of 64 (with 64^-0.25 pre-scale).
// One wave per row; two elements per lane; wave32 shuffle reductions.
// ---------------------------------------------------------------------------
__global__ __launch_bounds__(256) void softmax64(float* __restrict__ X) {
  const int wave = blockIdx.x * 8 + (threadIdx.x >> 5);   // 131072 rows
  const int lane = threadIdx.x & 31;
  const int m = wave >> 4;
  const int h = wave & 15;
  float* p = X + (size_t)m * DMODEL + h * DHEAD + 2 * lane;
  v2f v = *(v2f*)p;
  const float inv_scale = 0.3535533905932738f;            // 64^-0.25
  float a = v.x * inv_scale, c = v.y * inv_scale;
  float mx = fmaxf(a, c);
#pragma unroll
  for (int off = 16; off; off >>= 1) mx = fmaxf(mx, __shfl_xor(mx, off, 32));
  float e0 = __expf(a - mx), e1 = __expf(c - mx);
  float s = e0 + e1;
#pragma unroll
  for (int off = 16; off; off >>= 1) s += __shfl_xor(s, off, 32);
  float r = 1.0f / s;
  v.x = e0 * r;
  v.y = e1 * r;
  *(v2f*)p = v;
}

// ---------------------------------------------------------------------------
// A[b,h] = Ksm[b,h]^T @ V[b,h] : [64 x 2048] x [2048 x 64] -> [64 x 64]
// One block per (b,h); 16 waves, each a 16x16 tile; K = 2048.
// Result stored PACKED ( [(d>>1)*128 + e*2 + (d&1)] ) for bt_gemm's B loads.
// ---------------------------------------------------------------------------
__global__ __launch_bounds__(512) void kv_gemm(const float* __restrict__ Kbuf,
                                               const float* __restrict__ Vbuf,
                                               float* __restrict__ Abuf) {
  const int bh   = blockIdx.x;             // 0..63
  const int wave = threadIdx.x >> 5;       // 0..15
  const int lane = threadIdx.x & 31;
  const int lr   = lane & 15;
  const int khi  = (lane >> 4) << 1;
  const int m0   = (wave >> 2) << 4;       // d-tile
  const int n0   = (wave & 3) << 4;        // e-tile
  const int b = bh >> 4, h = bh & 15;

  const float* Kp = Kbuf + (size_t)b * SEQ * DMODEL + h * DHEAD;
  const float* Vp = Vbuf + (size_t)b * SEQ * DMODEL + h * DHEAD;
  const float* ka = Kp + m0 + lr;          // fixed d, stride over s
  const float* vb = Vp + n0 + lr;          // fixed e, stride over s

  v8f acc = {};
#pragma unroll 4
  for (int s = 0; s < SEQ; s += 4) {
    v2f a, bfr;
    a.x   = ka[(size_t)(s + khi) * DMODEL];
    a.y   = ka[(size_t)(s + khi + 1) * DMODEL];
    bfr.x = vb[(size_t)(s + khi) * DMODEL];
    bfr.y = vb[(size_t)(s + khi + 1) * DMODEL];
    acc = wmma_f32(a, bfr, acc);
  }

  float* Ap = Abuf + (size_t)bh * (DHEAD * DHEAD);
  const int ms = m0 + ((lane >> 4) << 3);
#pragma unroll
  for (int r = 0; r < 8; ++r) {
    int dd = ms + r;                        // k index of packed B
    Ap[(dd >> 1) * (2 * DHEAD) + (n0 + lr) * 2 + (dd & 1)] = acc[r];
  }
}

// ---------------------------------------------------------------------------
// Bt[b,h] = Qs[b,h] @ A[b,h] : [2048 x 64] x [64 x 64], A packed.
// One wave per 16(s) x 64(e) strip; output contiguous [b,h,s,d]
// (that flat buffer IS the torch .view reshape for the final GEMM).
// ---------------------------------------------------------------------------
__global__ __launch_bounds__(256) void bt_gemm(const float* __restrict__ Qbuf,
                                               const float* __restrict__ Abuf,
                                               float* __restrict__ Btbuf) {
  const int wave = blockIdx.x * 8 + (threadIdx.x >> 5);   // 8192 waves
  const int lane = threadIdx.x & 31;
  const int lr   = lane & 15;
  const int hi   = lane >> 4;
  const int khi  = hi << 1;
  const int bh = wave >> 7;                // 128 s-tiles per (b,h)
  const int m0 = (wave & 127) << 4;
  const int b = bh >> 4, h = bh & 15;

  const float* Qp = Qbuf + (size_t)b * SEQ * DMODEL + h * DHEAD;
  const float* Ap = Abuf + (size_t)bh * (DHEAD * DHEAD);
  const float* arow = Qp + (size_t)(m0 + lr) * DMODEL;
  const float* bp   = Ap + hi * (2 * DHEAD) + lr * 2;

  v8f acc0 = {}, acc1 = {}, acc2 = {}, acc3 = {};
#pragma unroll
  for (int k = 0; k < DHEAD; k += 4) {
    v2f a = *(const v2f*)(arow + k + khi);
    const float* bk = bp + (k >> 1) * (2 * DHEAD);
    v2f b0 = *(const v2f*)(bk + 0);
    v2f b1 = *(const v2f*)(bk + 32);
    v2f b2 = *(const v2f*)(bk + 64);
    v2f b3 = *(const v2f*)(bk + 96);
    acc0 = wmma_f32(a, b0, acc0);
    acc1 = wmma_f32(a, b1, acc1);
    acc2 = wmma_f32(a, b2, acc2);
    acc3 = wmma_f32(a, b3, acc3);
  }

  float* Btp = Btbuf + (size_t)bh * SEQ * DHEAD;
  const int ms = m0 + (hi << 3);
  float* cp = Btp + (size_t)ms * DHEAD + lr;
#pragma unroll
  for (int r = 0; r < 8; ++r) {
    cp[r * DHEAD +  0] = acc0[r];
    cp[r * DHEAD + 16] = acc1[r];
    cp[r * DHEAD + 32] = acc2[r];
    cp[r * DHEAD + 48] = acc3[r];
  }
}

// ---------------------------------------------------------------------------
extern "C" void kernel_launch(void* const* d_in, const int* in_sizes, int n_in,
                              void* d_out, int out_size, void* d_ws, size_t ws_size,
                              hipStream_t stream) {
  const float* x_q = (const float*)d_in[0];
  const float* x_k = (const float*)d_in[1];
  const float* x_v = (const float*)d_in[2];
  const float* W_Q = (const float*)d_in[3];
  const float* W_K = (const float*)d_in[4];
  const float* W_V = (const float*)d_in[5];
  const float* W_O = (const float*)d_in[6];
  float* out = (float*)d_out;

  // Workspace (floats): Q,K,V [8192x1024]; packed A [64*64*64];
  // Bt [8192x1024]; 4 packed weight panels [1024x1024] => ~152 MB
  float* ws    = (float*)d_ws;
  float* Q     = ws;
  float* K     = Q  + (size_t)MROWS * DMODEL;
  float* V     = K  + (size_t)MROWS * DMODEL;
  float* Abuf  = V  + (size_t)MROWS * DMODEL;
  float* Bt    = Abuf + (size_t)BQ * HEADS * DHEAD * DHEAD;
  float* Wq_p  = Bt  + (size_t)MROWS * DMODEL;
  float* Wk_p  = Wq_p + (size_t)DMODEL * DMODEL;
  float* Wv_p  = Wk_p + (size_t)DMODEL * DMODEL;
  float* Wo_p  = Wv_p + (size_t)DMODEL * DMODEL;

  // 1) pack all weight panels into WMMA-B interleaved layout (one-time, 16 MB)
  pack_w_head<<<4096, 256, 0, stream>>>(W_Q, Wq_p);
  pack_w_head<<<4096, 256, 0, stream>>>(W_K, Wk_p);
  pack_w_head<<<4096, 256, 0, stream>>>(W_V, Wv_p);
  pack_wo    <<<4096, 256, 0, stream>>>(W_O, Wo_p);

  // 2) Q/K/V projections: [8192,1024] x [1024,1024], f32 WMMA, b64-only loads
  gemm_mn<<<1024, 256, 0, stream>>>(x_q, Wq_p, Q);
  gemm_mn<<<1024, 256, 0, stream>>>(x_k, Wk_p, K);
  gemm_mn<<<1024, 256, 0, stream>>>(x_v, Wv_p, V);

  // 3) softmax over the 64 head-dims (64^-0.25 pre-scale), in place
  softmax64<<<16384, 256, 0, stream>>>(Q);
  softmax64<<<16384, 256, 0, stream>>>(K);

  // 4) A = Ksm^T @ V per (b,h), stored packed for bt_gemm
  kv_gemm<<<BQ * HEADS, 512, 0, stream>>>(K, V, Abuf);

  // 5) Bt = Qs @ A per (b,h), stored contiguous [b,h,s,d]
  bt_gemm<<<1024, 256, 0, stream>>>(Q, Abuf, Bt);

  // 6) out = Bt_flat @ W_O^T   (flat [b,h,s,d] == torch .view reshape)
  gemm_mn<<<1024, 256, 0, stream>>>(Bt, Wo_p, out);
}